// LongRangeFeaturizer_90443421319616
// MI455X (gfx1250) — compile-verified
//
#include <hip/hip_runtime.h>
#include <hip/hip_bf16.h>
#include <math.h>

// Problem constants (match reference)
#define BATCH 8
#define NATOM 2000
#define NATOMP 2048         // padded atoms for final kernel (writes guarded)
#define NEDGE 64000
#define DIM   256
#define KREAL 2196          // (2*6+1)^3 - 1
#define KPAD  2304          // padded: /16 tiles = 144 = 4 waves * 36 groups
#define RCUT_ 5.0f

// Workspace layout (floats)
#define QOFF   0u                                  // charges q: B*N*D
#define PROFF  (BATCH*NATOM*DIM)                   // pot_real:  B*N*D
#define GSCOFF (2u*BATCH*NATOM*DIM)                // G*Sc:      B*KPAD*D
#define GSSOFF (GSCOFF + BATCH*KPAD*DIM)           // G*Ss:      B*KPAD*D
#define KVOFF  (GSSOFF + BATCH*KPAD*DIM)           // kvec:      B*KPAD*3
#define GOFF   (KVOFF + BATCH*KPAD*3)              // G:         B*KPAD

typedef __attribute__((ext_vector_type(2))) float v2f;
typedef __attribute__((ext_vector_type(8))) float v8f;

// Native f32 WMMA: D(16x16) = A(16x4) * B(4x16) + C   (exact f32, matrix pipe)
__device__ __forceinline__ v8f wmma4(v2f a, v2f b, v8f c) {
  return __builtin_amdgcn_wmma_f32_16x16x4_f32(false, a, false, b, (short)0, c,
                                               false, false);
}

// Low 32 bits of a generic shared-memory pointer == LDS byte offset
// (AMDGPU addrspacecast: hi32 = LDS aperture, lo32 = workgroup LDS offset).
__device__ __forceinline__ unsigned lds_off(const void* p) {
  return (unsigned)(uintptr_t)p;
}

// Async global->LDS copy of 8 bytes (CDNA5 GLOBAL_LOAD_ASYNC_TO_LDS_B64,
// tracked by ASYNCcnt). ldsaddr: LDS byte offset; gp: global address.
__device__ __forceinline__ void async_copy_b64(unsigned ldsaddr, const float* gp) {
  asm volatile("global_load_async_to_lds_b64 %0, %1, off"
               :
               : "v"(ldsaddr), "v"(gp)
               : "memory");
}

__device__ __forceinline__ void wait_async() {
  asm volatile("s_wait_asynccnt 0x0" ::: "memory");
}

// ---------------------------------------------------------------------------
// 1) charges = features @ W^T + b     [16000 x 256] x [256 x 256]
// One wave per 16x16 output tile; K-loop in steps of 4 via f32 WMMA.
// ---------------------------------------------------------------------------
__global__ __launch_bounds__(128) void charges_gemm(
    const float* __restrict__ F, const float* __restrict__ W,
    const float* __restrict__ bias, float* __restrict__ q) {
  const int lane = threadIdx.x & 31;
  const int warp = threadIdx.x >> 5;
  const int tile = blockIdx.x * 4 + warp;            // 16000 tiles total
  const int tm   = tile >> 4;                        // 1000 row tiles
  const int tn   = tile & 15;                        // 16 col tiles
  const int l15  = lane & 15;
  const int hi   = lane >> 4;
  const int mrow = tm * 16 + l15;                    // A row for this lane
  const int ncol = tn * 16 + l15;                    // B col for this lane
  const int koff = hi * 2;

  v8f acc = {};
  for (int k0 = 0; k0 < DIM; k0 += 4) {
    v2f a, bf;
    a.x  = F[(size_t)mrow * DIM + k0 + koff];
    a.y  = F[(size_t)mrow * DIM + k0 + koff + 1];
    bf.x = W[(size_t)ncol * DIM + k0 + koff];        // B[k][n] = W[n][k]
    bf.y = W[(size_t)ncol * DIM + k0 + koff + 1];
    acc = wmma4(a, bf, acc);
  }
  const float bv = bias[ncol];
#pragma unroll
  for (int v = 0; v < 8; ++v) {
    int row = tm * 16 + v + 8 * hi;
    q[(size_t)row * DIM + ncol] = acc[v] + bv;
  }
}

// ---------------------------------------------------------------------------
// 2) Per-batch k-grid: kvec = NGRID @ (2*pi*inv(cell)^T), G = 4pi/V e^{-k2/2}/k2
// ---------------------------------------------------------------------------
__global__ __launch_bounds__(256) void kgrid_kernel(
    const float* __restrict__ cell, float* __restrict__ kvec,
    float* __restrict__ gfac) {
  const int b = blockIdx.x;
  const float* c = cell + (size_t)b * 9;
  const float a00=c[0],a01=c[1],a02=c[2],a10=c[3],a11=c[4],a12=c[5],
              a20=c[6],a21=c[7],a22=c[8];
  const float det = a00*(a11*a22-a12*a21) - a01*(a10*a22-a12*a20)
                  + a02*(a10*a21-a11*a20);
  const float id = 1.0f / det;
  const float i00=(a11*a22-a12*a21)*id, i01=(a02*a21-a01*a22)*id, i02=(a01*a12-a02*a11)*id;
  const float i10=(a12*a20-a10*a22)*id, i11=(a00*a22-a02*a20)*id, i12=(a02*a10-a00*a12)*id;
  const float i20=(a10*a21-a11*a20)*id, i21=(a01*a20-a00*a21)*id, i22=(a00*a11-a01*a10)*id;
  const float twopi = 6.283185307179586f;
  const float gpre  = 4.0f * 3.14159265358979f / fabsf(det);

  for (int k = threadIdx.x; k < KPAD; k += blockDim.x) {
    float kx = 0.f, ky = 0.f, kz = 0.f, g = 0.f;
    if (k < KREAL) {
      const int idx = (k < 1098) ? k : k + 1;        // skip (0,0,0) at flat 1098
      const float ni = (float)(idx / 169 - 6);
      const float nj = (float)((idx / 13) % 13 - 6);
      const float nk = (float)(idx % 13 - 6);
      kx = twopi * (ni * i00 + nj * i01 + nk * i02); // sum_j n_j * inv[c][j]
      ky = twopi * (ni * i10 + nj * i11 + nk * i12);
      kz = twopi * (ni * i20 + nj * i21 + nk * i22);
      const float k2 = kx * kx + ky * ky + kz * kz;
      g = gpre * __expf(-0.5f * k2) / k2;            // sigma = 1
    }
    kvec[((size_t)b * KPAD + k) * 3 + 0] = kx;
    kvec[((size_t)b * KPAD + k) * 3 + 1] = ky;
    kvec[((size_t)b * KPAD + k) * 3 + 2] = kz;
    gfac[(size_t)b * KPAD + k] = g;
  }
}

// ---------------------------------------------------------------------------
// 3) Real-space: pot_real[i,:] += v_sr(r_e) * q[j,:]  (one wave per edge)
// ---------------------------------------------------------------------------
__global__ __launch_bounds__(256) void edge_real(
    const int* __restrict__ nbidx, const float* __restrict__ nbdist,
    const float* __restrict__ q, float* __restrict__ preal) {
  const int lane = threadIdx.x & 31;
  const int warp = threadIdx.x >> 5;
  const long long eg = (long long)blockIdx.x * 8 + warp;
  const int b = (int)(eg / NEDGE);
  const int e = (int)(eg % NEDGE);
  const int i = nbidx[((size_t)b * NEDGE + e) * 2 + 0];
  const int j = nbidx[((size_t)b * NEDGE + e) * 2 + 1];
  const float r = nbdist[(size_t)b * NEDGE + e];
  const float inv_s = 0.7071067811865475f;           // 1/(sqrt(2)*sigma)
  const float v = erfcf(r * inv_s) / r;
  const float fcut =
      (r < RCUT_) ? 0.5f * (__cosf(3.14159265358979f * r / RCUT_) + 1.0f) : 0.0f;
  const float vsr = v - fcut / r;
  const float* qj = q + ((size_t)b * NATOM + j) * DIM;
  float* pi = preal + ((size_t)b * NATOM + i) * DIM;
#pragma unroll
  for (int t = 0; t < 8; ++t) {
    const int d = lane + 32 * t;
    atomicAdd(&pi[d], vsr * qj[d]);
  }
}

// ---------------------------------------------------------------------------
// 4) Structure factors: GSc[k,d] = G[k] * sum_n cos(pos_n . k_k) q[n,d]
//    (GSs with sin). Block = 4 waves = 4 k-tiles sharing one d-tile.
//    q chunks (16 atoms x 16 d) staged into LDS with async global->LDS B64
//    (one per thread), consumed by all 4 waves -> 4x less global B traffic.
// ---------------------------------------------------------------------------
__global__ __launch_bounds__(128) void structure_factor(
    const float* __restrict__ pos, const float* __restrict__ kvec,
    const float* __restrict__ gfac, const float* __restrict__ q,
    float* __restrict__ gsc, float* __restrict__ gss) {
  __shared__ __align__(16) float lq[16][16];         // staged q chunk

  const int tid  = threadIdx.x;
  const int lane = tid & 31;
  const int warp = tid >> 5;
  const int bi   = blockIdx.x;                       // 8 * 36 * 16 blocks
  const int b    = bi / ((KPAD / 64) * 16);          // KPAD/64 = 36 k-groups
  const int rem  = bi % ((KPAD / 64) * 16);
  const int kg   = rem >> 4;                         // k-tile group (4 tiles)
  const int dt   = rem & 15;
  const int kt   = kg * 4 + warp;                    // this wave's k-tile
  const int l15  = lane & 15;
  const int hi   = lane >> 4;

  const int krow = kt * 16 + l15;                    // A-matrix row (k index)
  const float kx = kvec[((size_t)b * KPAD + krow) * 3 + 0];
  const float ky = kvec[((size_t)b * KPAD + krow) * 3 + 1];
  const float kz = kvec[((size_t)b * KPAD + krow) * 3 + 2];
  const int d0 = dt * 16;
  const float* qb = q + (size_t)b * NATOM * DIM;
  const float* pb = pos + (size_t)b * NATOM * 3;

  // staging pattern: thread t copies q[n0 + (t>>3)][d0 + 2*(t&7) .. +1]
  const int sr = tid >> 3;
  const int scl = (tid & 7) * 2;
  const unsigned ldst = lds_off(&lq[sr][scl]);

  v8f accC = {}, accS = {};
  for (int n0 = 0; n0 < NATOM; n0 += 16) {
    async_copy_b64(ldst, qb + (size_t)(n0 + sr) * DIM + d0 + scl);
    wait_async();
    __syncthreads();
#pragma unroll
    for (int ks = 0; ks < 16; ks += 4) {
      const int a0 = ks + hi * 2;                    // local atom pair
      const int ga = n0 + a0;
      const float ph0 = pb[ga*3+0]*kx + pb[ga*3+1]*ky + pb[ga*3+2]*kz;
      const float ph1 = pb[ga*3+3]*kx + pb[ga*3+4]*ky + pb[ga*3+5]*kz;
      float c0, s0, c1, s1;
      __sincosf(ph0, &s0, &c0);
      __sincosf(ph1, &s1, &c1);
      v2f aC, aS, bq;
      aC.x = c0; aC.y = c1;
      aS.x = s0; aS.y = s1;
      bq.x = lq[a0][l15];
      bq.y = lq[a0 + 1][l15];
      accC = wmma4(aC, bq, accC);
      accS = wmma4(aS, bq, accS);
    }
    __syncthreads();
  }
#pragma unroll
  for (int v = 0; v < 8; ++v) {
    const int kr = kt * 16 + v + 8 * hi;
    const float g = gfac[(size_t)b * KPAD + kr];     // 0 on padded rows
    gsc[((size_t)b * KPAD + kr) * DIM + d0 + l15] = accC[v] * g;
    gss[((size_t)b * KPAD + kr) * DIM + d0 + l15] = accS[v] * g;
  }
}

// ---------------------------------------------------------------------------
// 5) pot_recip[n,d] = sum_k cos(ph)*GSc + sin(ph)*GSs, fused epilogue
//    out = (pot_real + pot_recip + self*q) * q.
//    Block = 4 waves = 4 atom-tiles sharing one d-tile; GSc/GSs chunks
//    (16 k x 16 d) staged via async global->LDS B64, shared by all waves.
//    Atoms padded to 2048: position reads clamped, writes guarded after all
//    WMMAs so EXEC is all-ones at every matrix op.
// ---------------------------------------------------------------------------
__global__ __launch_bounds__(128) void recip_final(
    const float* __restrict__ pos, const float* __restrict__ kvec,
    const float* __restrict__ q, const float* __restrict__ preal,
    const float* __restrict__ gsc, const float* __restrict__ gss,
    float* __restrict__ out) {
  __shared__ __align__(16) float lsc[16][16];
  __shared__ __align__(16) float lss[16][16];

  const int tid  = threadIdx.x;
  const int lane = tid & 31;
  const int warp = tid >> 5;
  const int bi   = blockIdx.x;                       // 8 * 32 * 16 blocks
  const int b    = bi / ((NATOMP / 64) * 16);        // NATOMP/64 = 32 groups
  const int rem  = bi % ((NATOMP / 64) * 16);
  const int ag   = rem >> 4;                         // atom-tile group
  const int dt   = rem & 15;
  const int at   = ag * 4 + warp;                    // this wave's atom tile
  const int l15  = lane & 15;
  const int hi   = lane >> 4;

  const int atomA = at * 16 + l15;                   // A-matrix row (atom)
  const int atomC = atomA < NATOM ? atomA : NATOM - 1;  // clamped for loads
  const float* pb = pos + (size_t)b * NATOM * 3;
  const float px = pb[atomC*3+0], py = pb[atomC*3+1], pz = pb[atomC*3+2];
  const int d0 = dt * 16;
  const float* kvb  = kvec + (size_t)b * KPAD * 3;
  const float* gscb = gsc + (size_t)b * KPAD * DIM;
  const float* gssb = gss + (size_t)b * KPAD * DIM;

  const int sr  = tid >> 3;                          // staging row (k)
  const int scl = (tid & 7) * 2;                     // staging col pair (d)
  const unsigned ldsc = lds_off(&lsc[sr][scl]);
  const unsigned ldss = lds_off(&lss[sr][scl]);

  v8f acc = {};
  for (int k0 = 0; k0 < KPAD; k0 += 16) {
    async_copy_b64(ldsc, gscb + (size_t)(k0 + sr) * DIM + d0 + scl);
    async_copy_b64(ldss, gssb + (size_t)(k0 + sr) * DIM + d0 + scl);
    wait_async();
    __syncthreads();
#pragma unroll
    for (int ks = 0; ks < 16; ks += 4) {
      const int kl = ks + hi * 2;                    // local k pair
      const int kg = k0 + kl;
      const float ph0 = px*kvb[kg*3+0] + py*kvb[kg*3+1] + pz*kvb[kg*3+2];
      const float ph1 = px*kvb[kg*3+3] + py*kvb[kg*3+4] + pz*kvb[kg*3+5];
      float c0, s0, c1, s1;
      __sincosf(ph0, &s0, &c0);
      __sincosf(ph1, &s1, &c1);
      v2f aC, aS, bC, bS;
      aC.x = c0; aC.y = c1;
      aS.x = s0; aS.y = s1;
      bC.x = lsc[kl][l15];  bC.y = lsc[kl + 1][l15];
      bS.x = lss[kl][l15];  bS.y = lss[kl + 1][l15];
      acc = wmma4(aC, bC, acc);
      acc = wmma4(aS, bS, acc);
    }
    __syncthreads();
  }
  const float selfc = -0.7978845608028654f;          // -sqrt(2/pi)/sigma
#pragma unroll
  for (int v = 0; v < 8; ++v) {
    const int atom = at * 16 + v + 8 * hi;
    if (atom < NATOM) {
      const size_t idx = ((size_t)b * NATOM + atom) * DIM + d0 + l15;
      const float qv = q[idx];
      out[idx] = (acc[v] + preal[idx] + selfc * qv) * qv;
    }
  }
}

// ---------------------------------------------------------------------------
extern "C" void kernel_launch(void* const* d_in, const int* in_sizes, int n_in,
                              void* d_out, int out_size, void* d_ws,
                              size_t ws_size, hipStream_t stream) {
  (void)in_sizes; (void)n_in; (void)out_size; (void)ws_size;
  const float* features  = (const float*)d_in[0];
  const float* positions = (const float*)d_in[1];
  const float* cell      = (const float*)d_in[2];
  const int*   nbidx     = (const int*)d_in[3];
  const float* nbdist    = (const float*)d_in[4];
  const float* W         = (const float*)d_in[5];
  const float* bias      = (const float*)d_in[6];
  float* out = (float*)d_out;
  float* ws  = (float*)d_ws;

  float* q     = ws + QOFF;
  float* preal = ws + PROFF;
  float* gsc   = ws + GSCOFF;
  float* gss   = ws + GSSOFF;
  float* kvec  = ws + KVOFF;
  float* gfac  = ws + GOFF;

  // pot_real accumulates via atomics -> must be zeroed every call
  hipMemsetAsync(preal, 0, (size_t)BATCH * NATOM * DIM * sizeof(float), stream);

  charges_gemm<<<(BATCH * NATOM / 16) * (DIM / 16) / 4, 128, 0, stream>>>(
      features, W, bias, q);
  kgrid_kernel<<<BATCH, 256, 0, stream>>>(cell, kvec, gfac);
  edge_real<<<BATCH * NEDGE / 8, 256, 0, stream>>>(nbidx, nbdist, q, preal);
  structure_factor<<<BATCH * (KPAD / 64) * 16, 128, 0, stream>>>(
      positions, kvec, gfac, q, gsc, gss);
  recip_final<<<BATCH * (NATOMP / 64) * 16, 128, 0, stream>>>(
      positions, kvec, q, preal, gsc, gss, out);
}